// Skip_1_34832184770675
// MI455X (gfx1250) — compile-verified
//
#include <hip/hip_runtime.h>
#include <stdint.h>

typedef __attribute__((ext_vector_type(2))) float v2f;
typedef __attribute__((ext_vector_type(8))) float v8f;

// Padded LDS row stride (floats): multiple of 4 (16B-aligned rows for b128),
// 68 mod 64-banks = 4 => transposed accesses hit distinct banks.
#define LD 68

#define WMMA_F32(A, B, C) \
  __builtin_amdgcn_wmma_f32_16x16x4_f32(false, (A), false, (B), (short)0, (C), false, false)

// out[b,o] = sum_c Wc @ x[b,c] @ Wc^T,  Wc = W[o, c>>1, :, (c&1)*64 : +64]
// computed as T' = x @ Wc^T (stored transposed in LDS), then out += Wc @ T'.
// Channel loop is software-pipelined: async DMA for channel c+1 overlaps the
// two WMMA passes of channel c (double-buffered sX/sW).
__global__ __launch_bounds__(256, 2)
void bimap_spd_kernel(const float* __restrict__ x,
                      const float* __restrict__ W,
                      float* __restrict__ out) {
  __shared__ float sX[2][64 * LD];   // x[b,c], double-buffered
  __shared__ float sW[2][64 * LD];   // Wc[o,c], double-buffered
  __shared__ float sTt[64 * LD];     // (x @ Wc^T)^T

  const int bo = blockIdx.x;  // 0..2047
  const int b  = bo >> 3;
  const int o  = bo & 7;

  const int tid  = threadIdx.x;
  const int lane = tid & 31;
  const int wave = tid >> 5;              // 0..7
  const int m0   = (wave >> 1) << 4;      // row band (0,16,32,48)
  const int n0   = (wave & 1) << 5;       // col half (0 or 32); tiles n0, n0+16

  const int l15   = lane & 15;
  const int lhalf = lane >> 4;
  const int koff  = lhalf << 1;           // A/B fragment K sub-offset
  const int moff  = lhalf << 3;           // C/D fragment M offset

  // Issue one channel's staging as 8 async b128 DMAs (16B/lane each).
  auto stage = [&](int c, int buf) __attribute__((always_inline)) {
    const int p = c >> 1;
    const int h = c & 1;
    const float* xsrc = x + (size_t)(b * 16 + c) * (64 * 64);
    const float* wsrc = W + (size_t)(o * 8 + p) * (64 * 128) + h * 64;
#pragma unroll
    for (int j = 0; j < 4; ++j) {
      const int i = tid + j * 256;
      const int r = i >> 4;         // source row
      const int q = (i & 15) << 2;  // float offset within row
      const uint32_t lx = (uint32_t)(uintptr_t)(&sX[buf][r * LD + q]);
      const float* gx = xsrc + r * 64 + q;
      asm volatile("global_load_async_to_lds_b128 %0, %1, off"
                   :: "v"(lx), "v"(gx) : "memory");
      const uint32_t lw = (uint32_t)(uintptr_t)(&sW[buf][r * LD + q]);
      const float* gw = wsrc + (size_t)r * 128 + q;
      asm volatile("global_load_async_to_lds_b128 %0, %1, off"
                   :: "v"(lw), "v"(gw) : "memory");
    }
  };

  // 4 independent accumulator chains: {tile0, tile1} x {even-k, odd-k}
  v8f acc0a = {}, acc0b = {}, acc1a = {}, acc1b = {};

  stage(0, 0);  // prologue: channel 0 DMA in flight

  for (int c = 0; c < 16; ++c) {
    const int cur = c & 1;
    const float* __restrict__ cX = sX[cur];
    const float* __restrict__ cW = sW[cur];

    // Drain own DMAs for buf[cur]; barrier publishes them to all waves AND
    // guarantees everyone is done reading buf[cur^1] (prev step 2) before we
    // overwrite it below.
    asm volatile("s_wait_asynccnt 0x0" ::: "memory");
    __syncthreads();

    if (c < 15) stage(c + 1, cur ^ 1);  // overlap next DMA with this compute

    // ---- Step 1: T' = x @ Wc^T  (A = x rows, B[k][n] = Wc[n][k]; both
    // fragments contiguous b64 LDS reads). 4 chains for ILP.
    v8f t0a = {}, t0b = {}, t1a = {}, t1b = {};
#pragma unroll
    for (int k0 = 0; k0 < 64; k0 += 8) {
      v2f a, b0, b1;
      const float* ar  = cX + (m0 + l15) * LD + (k0 + koff);
      const float* br0 = cW + (n0 + l15) * LD + (k0 + koff);
      const float* br1 = cW + (n0 + 16 + l15) * LD + (k0 + koff);
      a.x = ar[0];   a.y = ar[1];
      b0.x = br0[0]; b0.y = br0[1];
      b1.x = br1[0]; b1.y = br1[1];
      t0a = WMMA_F32(a, b0, t0a);
      t1a = WMMA_F32(a, b1, t1a);
      a.x = ar[4];   a.y = ar[5];
      b0.x = br0[4]; b0.y = br0[5];
      b1.x = br1[4]; b1.y = br1[5];
      t0b = WMMA_F32(a, b0, t0b);
      t1b = WMMA_F32(a, b1, t1b);
    }
    {
      const v8f t0 = t0a + t0b;
      const v8f t1 = t1a + t1b;
      // Store transposed: sTt[n][m] = T'[m][n]; per lane the 8 C/D elements
      // span consecutive m -> two contiguous b128 stores.
      float* d0 = sTt + (n0 + l15) * LD + (m0 + moff);
      float* d1 = sTt + (n0 + 16 + l15) * LD + (m0 + moff);
      ((float4*)d0)[0] = make_float4(t0[0], t0[1], t0[2], t0[3]);
      ((float4*)d0)[1] = make_float4(t0[4], t0[5], t0[6], t0[7]);
      ((float4*)d1)[0] = make_float4(t1[0], t1[1], t1[2], t1[3]);
      ((float4*)d1)[1] = make_float4(t1[4], t1[5], t1[6], t1[7]);
    }
    __syncthreads();  // sTt visible to all waves

    // ---- Step 2: acc += Wc @ T'  (A = Wc rows, B[k][n] = T'[k][n] = sTt[n][k];
    // all-contiguous b64 LDS reads).
#pragma unroll
    for (int k0 = 0; k0 < 64; k0 += 8) {
      v2f a, b0, b1;
      const float* ar  = cW + (m0 + l15) * LD + (k0 + koff);
      const float* br0 = sTt + (n0 + l15) * LD + (k0 + koff);
      const float* br1 = sTt + (n0 + 16 + l15) * LD + (k0 + koff);
      a.x = ar[0];   a.y = ar[1];
      b0.x = br0[0]; b0.y = br0[1];
      b1.x = br1[0]; b1.y = br1[1];
      acc0a = WMMA_F32(a, b0, acc0a);
      acc1a = WMMA_F32(a, b1, acc1a);
      a.x = ar[4];   a.y = ar[5];
      b0.x = br0[4]; b0.y = br0[5];
      b1.x = br1[4]; b1.y = br1[5];
      acc0b = WMMA_F32(a, b0, acc0b);
      acc1b = WMMA_F32(a, b1, acc1b);
    }
  }

  const v8f acc0 = acc0a + acc0b;
  const v8f acc1 = acc1a + acc1b;
  float* dst = out + (size_t)(b * 8 + o) * (64 * 64);
#pragma unroll
  for (int v = 0; v < 8; ++v) {
    dst[(m0 + moff + v) * 64 + (n0 + l15)] = acc0[v];
    dst[(m0 + moff + v) * 64 + (n0 + 16 + l15)] = acc1[v];
  }
}

extern "C" void kernel_launch(void* const* d_in, const int* in_sizes, int n_in,
                              void* d_out, int out_size, void* d_ws, size_t ws_size,
                              hipStream_t stream) {
  const float* x = (const float*)d_in[0];  // (256, 16, 64, 64) f32
  const float* W = (const float*)d_in[1];  // (8, 8, 64, 128) f32
  float* out = (float*)d_out;              // (256, 8, 64, 64) f32

  dim3 grid(256 * 8);  // one workgroup per (b, o)
  dim3 block(256);     // 8 wave32, 2 output tiles per wave
  hipLaunchKernelGGL(bimap_spd_kernel, grid, block, 0, stream, x, W, out);
}